// Attn_75076028334217
// MI455X (gfx1250) — compile-verified
//
#include <hip/hip_runtime.h>

#define L_DIM 2048
#define B_DIM 32
#define H_DIM 1024
#define FLOAT_MIN_V (-1e20f)

typedef float v2f __attribute__((ext_vector_type(2)));
typedef float v4f __attribute__((ext_vector_type(4)));
typedef float v8f __attribute__((ext_vector_type(8)));

__device__ __forceinline__ float waveSum(float x) {
#pragma unroll
  for (int off = 16; off > 0; off >>= 1) x += __shfl_xor(x, off, 32);
  return x;
}

__device__ __forceinline__ float waveMax(float x) {
#pragma unroll
  for (int off = 16; off > 0; off >>= 1) x = fmaxf(x, __shfl_xor(x, off, 32));
  return x;
}

// ---------------------------------------------------------------------------
// Kernel A: u[b,h] = sum_g hidden[b,g] * W[g,h]   (32x1024 @ 1024x1024)
// One wave per 16(b) x 16(h) tile of u, K stepped by 4 via
// V_WMMA_F32_16X16X4_F32.  A (16x4 f32): lanes 0-15 hold M=lane with
// K=0 (vgpr0) / K=1 (vgpr1); lanes 16-31 hold K=2/3.  B (4x16): row K
// striped across lanes within a VGPR (same half-wave split).
// C/D (16x16 f32): vgpr i -> M = i + 8*(lane>=16), N = lane%16.
// ---------------------------------------------------------------------------
__global__ void __launch_bounds__(32)
hiddenW_wmma(const float* __restrict__ hidden,   // [B,H]
             const float* __restrict__ W,        // [H(g), H(h)]
             float* __restrict__ u) {            // [B,H]
  const int lane = threadIdx.x;
  const int row  = lane & 15;       // M for A-tile, N for B/D tiles
  const int hi   = lane >> 4;       // half-wave: selects K pair 0/1 vs 2/3
  const int hbase = blockIdx.x * 16;
  const int bbase = blockIdx.y * 16;

  v8f acc = {};
  for (int k = 0; k < H_DIM; k += 4) {
    const int ka = k + 2 * hi;
    v2f a, bm;
    a.x  = hidden[(bbase + row) * H_DIM + ka];
    a.y  = hidden[(bbase + row) * H_DIM + ka + 1];
    bm.x = W[(size_t)ka * H_DIM + hbase + row];
    bm.y = W[(size_t)(ka + 1) * H_DIM + hbase + row];
    acc = __builtin_amdgcn_wmma_f32_16x16x4_f32(
        /*neg_a=*/false, a, /*neg_b=*/false, bm,
        /*c_mod=*/(short)0, acc, /*reuse_a=*/false, /*reuse_b=*/false);
  }
#pragma unroll
  for (int i = 0; i < 8; ++i) {
    const int m = i + 8 * hi;
    u[(bbase + m) * H_DIM + hbase + row] = acc[i];
  }
}

// ---------------------------------------------------------------------------
// Kernel B: c[b] = hidden[b,:] . bias
// ---------------------------------------------------------------------------
__global__ void __launch_bounds__(256)
bias_dot(const float* __restrict__ hidden, const float* __restrict__ bias,
         float* __restrict__ c) {
  const int b = blockIdx.x, tid = threadIdx.x;
  float s = 0.f;
  for (int g = tid; g < H_DIM; g += 256)
    s = fmaf(hidden[b * H_DIM + g], bias[g], s);
  s = waveSum(s);
  __shared__ float sm[8];
  if ((tid & 31) == 0) sm[tid >> 5] = s;
  __syncthreads();
  if (tid == 0) {
    float t = 0.f;
#pragma unroll
    for (int i = 0; i < 8; ++i) t += sm[i];
    c[b] = t;
  }
}

// ---------------------------------------------------------------------------
// Kernel C: energy[b,l] = enc[l,b,:] . u[b,:] + c[b]  (masked)
// One wave per (l,b); enc streamed once (256 MB) with NT hint -> HBM bound.
// ---------------------------------------------------------------------------
__global__ void __launch_bounds__(256)
energies_kernel(const float* __restrict__ enc,      // [L,B,H]
                const float* __restrict__ u,        // [B,H]
                const float* __restrict__ c,        // [B]
                const int* __restrict__ lengths,    // [B]
                float* __restrict__ energies) {     // [B,L]
  const int wid  = (blockIdx.x * blockDim.x + threadIdx.x) >> 5; // = l*B + b
  const int lane = threadIdx.x & 31;
  const int l = wid >> 5;       // / B_DIM
  const int b = wid & (B_DIM - 1);

  const v4f* ev = (const v4f*)(enc + (size_t)wid * H_DIM);
  const v4f* uv = (const v4f*)(u + (size_t)b * H_DIM);

  float acc = 0.f;
#pragma unroll
  for (int i = 0; i < 8; ++i) {
    v4f e = __builtin_nontemporal_load(&ev[i * 32 + lane]); // read-once stream
    v4f w = uv[i * 32 + lane];                              // hot in L2/L0
    acc = fmaf(e.x, w.x, acc);
    acc = fmaf(e.y, w.y, acc);
    acc = fmaf(e.z, w.z, acc);
    acc = fmaf(e.w, w.w, acc);
  }
  acc = waveSum(acc);
  if (lane == 0) {
    const float e = acc + c[b];
    energies[b * L_DIM + l] = (l < lengths[b]) ? e : FLOAT_MIN_V;
  }
}

// ---------------------------------------------------------------------------
// Kernel D: masked softmax + renormalize over valid positions, per b.
// out[b,l] = valid ? exp(e-max)/sum_valid exp(e-max) : 0
// ---------------------------------------------------------------------------
__global__ void __launch_bounds__(256)
softmax_kernel(const float* __restrict__ energies,  // [B,L]
               const int* __restrict__ lengths,
               float* __restrict__ out) {           // [B,1,L]
  const int b = blockIdx.x, tid = threadIdx.x;
  const int lane = tid & 31, wave = tid >> 5;
  const int len = lengths[b];

  float v[8];
  float m = -3.0e38f;
#pragma unroll
  for (int i = 0; i < 8; ++i) {
    v[i] = energies[b * L_DIM + tid + i * 256];
    m = fmaxf(m, v[i]);
  }
  m = waveMax(m);
  __shared__ float smx[8], ssm[8];
  if (lane == 0) smx[wave] = m;
  __syncthreads();
#pragma unroll
  for (int i = 0; i < 8; ++i) m = fmaxf(m, smx[i]);

  float s = 0.f;
#pragma unroll
  for (int i = 0; i < 8; ++i) {
    const int l = tid + i * 256;
    const float e = (l < len) ? __expf(v[i] - m) : 0.f;
    v[i] = e;
    s += e;
  }
  s = waveSum(s);
  if (lane == 0) ssm[wave] = s;
  __syncthreads();
  s = 0.f;
#pragma unroll
  for (int i = 0; i < 8; ++i) s += ssm[i];

  const float inv = 1.0f / s;
#pragma unroll
  for (int i = 0; i < 8; ++i)
    out[b * L_DIM + tid + i * 256] = v[i] * inv;
}

// ---------------------------------------------------------------------------
extern "C" void kernel_launch(void* const* d_in, const int* in_sizes, int n_in,
                              void* d_out, int out_size, void* d_ws, size_t ws_size,
                              hipStream_t stream) {
  const float* hidden  = (const float*)d_in[0];   // [1,B,H]
  const float* enc     = (const float*)d_in[1];   // [L,B,H]
  const float* W       = (const float*)d_in[2];   // [H,H]
  const float* bias    = (const float*)d_in[3];   // [H]
  const int*   lengths = (const int*)d_in[4];     // [B]
  float* out = (float*)d_out;                     // [B,1,L]

  float* u        = (float*)d_ws;                 // B*H floats
  float* c        = u + B_DIM * H_DIM;            // B floats
  float* energies = c + 32;                       // B*L floats

  hiddenW_wmma<<<dim3(H_DIM / 16, B_DIM / 16), 32, 0, stream>>>(hidden, W, u);
  bias_dot<<<B_DIM, 256, 0, stream>>>(hidden, bias, c);
  energies_kernel<<<(L_DIM * B_DIM) / 8, 256, 0, stream>>>(enc, u, c, lengths, energies);
  softmax_kernel<<<B_DIM, 256, 0, stream>>>(energies, lengths, out);
}